// ClusteringLoss_48009144434956
// MI455X (gfx1250) — compile-verified
//
#include <hip/hip_runtime.h>
#include <math.h>

// Problem constants from the reference
#define BB   4
#define NBP  131072
#define KK   32
#define NTOT (BB * NBP)
#define HALF_S 384.0f

// Workspace float offsets
#define WS_SUMS   0      // B*K*5: cnt, sx, sy, sz, ssig   (atomic)
#define WS_ACC    640    // B*4: mask, smooth, seed, acchits (atomic)
#define WS_CEN    656    // B*K*3 centers
#define WS_I2S    1040   // B*K  1/(2*sigma^2)
#define WS_CN2    1168   // B*K  ||center||^2
#define WS_SGK    1296   // B*K  sigma_k
#define WS_INV    1424   // B*K  1/counts
#define WS_TOTAL  1552

typedef __attribute__((ext_vector_type(2))) float v2f;
typedef __attribute__((ext_vector_type(8))) float v8f;

__global__ __launch_bounds__(256) void cl_zero(float* ws) {
  int i = blockIdx.x * 256 + threadIdx.x;
  if (i < WS_TOTAL) ws[i] = 0.0f;
}

// Pass 1: per-(b,k) sums via LDS atomics -> global atomics.
// Each 256-thread block covers 256 contiguous points (one batch; NB % 256 == 0).
__global__ __launch_bounds__(256) void cl_pass1(const float* __restrict__ le,
                                                const float* __restrict__ lab,
                                                float* __restrict__ ws) {
  __shared__ float bins[KK * 5];
  const int tid = threadIdx.x;
  if (tid < KK * 5) bins[tid] = 0.0f;
  __syncthreads();
  const int n = blockIdx.x * 256 + tid;
  const int b = n / NBP;
  const float cex = le[n * 4 + 0] + (lab[n * 6 + 1] - HALF_S) / HALF_S;
  const float cey = le[n * 4 + 1] + (lab[n * 6 + 2] - HALF_S) / HALF_S;
  const float cez = le[n * 4 + 2] + (lab[n * 6 + 3] - HALF_S) / HALF_S;
  const float sig = le[n * 4 + 3];
  const int   k   = (int)lab[n * 6 + 5];
  atomicAdd(&bins[k * 5 + 0], 1.0f);
  atomicAdd(&bins[k * 5 + 1], cex);
  atomicAdd(&bins[k * 5 + 2], cey);
  atomicAdd(&bins[k * 5 + 3], cez);
  atomicAdd(&bins[k * 5 + 4], sig);
  __syncthreads();
  if (tid < KK * 5) atomicAdd(&ws[WS_SUMS + b * KK * 5 + tid], bins[tid]);
}

// Finalize the 128 cluster stats.
__global__ __launch_bounds__(128) void cl_finalize(float* __restrict__ ws) {
  const int t = threadIdx.x;   // t = b*K + k, 0..127
  if (t >= BB * KK) return;
  const float* s = ws + WS_SUMS + t * 5;
  const float c  = fmaxf(s[0], 1.0f);
  const float ic = 1.0f / c;
  const float cx = s[1] * ic, cy = s[2] * ic, cz = s[3] * ic, sg = s[4] * ic;
  ws[WS_CEN + t * 3 + 0] = cx;
  ws[WS_CEN + t * 3 + 1] = cy;
  ws[WS_CEN + t * 3 + 2] = cz;
  ws[WS_I2S + t] = 1.0f / (2.0f * sg * sg);
  ws[WS_CN2 + t] = cx * cx + cy * cy + cz * cz;
  ws[WS_SGK + t] = sg;
  ws[WS_INV + t] = ic;
}

// Pass 2: WMMA f32 16x16x4 computes ce . center^T for 32 points x 32 clusters
// per wave; scalar epilogue does clip/exp/log1p/argmin and per-batch reduction.
__global__ __launch_bounds__(256) void cl_pass2(const float* __restrict__ le,
                                                const float* __restrict__ se,
                                                const float* __restrict__ lab,
                                                float* __restrict__ ws) {
  __shared__ float s_cx[KK], s_cy[KK], s_cz[KK];
  __shared__ float s_i2[KK], s_n2[KK], s_sg[KK], s_ic[KK];
  __shared__ float s_ce[8][KK][4];     // [wave][point-in-wave][xyz_]
  __shared__ float s_dot[8][KK][33];   // [wave][point][cluster], padded pitch
  __shared__ float s_acc[4];

  const int tid = threadIdx.x;
  const int w = tid >> 5;
  const int l = tid & 31;
  const int n = blockIdx.x * 256 + tid;   // all points in block share batch b
  const int b = n / NBP;
  const int bk = b * KK;

  if (tid < KK) {
    s_cx[tid] = ws[WS_CEN + (bk + tid) * 3 + 0];
    s_cy[tid] = ws[WS_CEN + (bk + tid) * 3 + 1];
    s_cz[tid] = ws[WS_CEN + (bk + tid) * 3 + 2];
    s_i2[tid] = ws[WS_I2S + bk + tid];
    s_n2[tid] = ws[WS_CN2 + bk + tid];
    s_sg[tid] = ws[WS_SGK + bk + tid];
    s_ic[tid] = ws[WS_INV + bk + tid];
  }
  if (tid < 4) s_acc[tid] = 0.0f;

  const float cex = le[n * 4 + 0] + (lab[n * 6 + 1] - HALF_S) / HALF_S;
  const float cey = le[n * 4 + 1] + (lab[n * 6 + 2] - HALF_S) / HALF_S;
  const float cez = le[n * 4 + 2] + (lab[n * 6 + 3] - HALF_S) / HALF_S;
  const float sig = le[n * 4 + 3];
  const float sd  = se[n];
  const int  inst = (int)lab[n * 6 + 5];
  s_ce[w][l][0] = cex; s_ce[w][l][1] = cey; s_ce[w][l][2] = cez;
  __syncthreads();

  // Build WMMA fragments (ISA 16x4 f32 A layout: lanes 0-15 -> K=0,1; 16-31 -> K=2,3)
  const int  m  = l & 15;
  const bool hi = (l >= 16);
  v2f a0, a1, b0, b1;
  a0.x = hi ? s_ce[w][m][2]      : s_ce[w][m][0];
  a0.y = hi ? 0.0f               : s_ce[w][m][1];
  a1.x = hi ? s_ce[w][m + 16][2] : s_ce[w][m + 16][0];
  a1.y = hi ? 0.0f               : s_ce[w][m + 16][1];
  b0.x = hi ? s_cz[m]            : s_cx[m];
  b0.y = hi ? 0.0f               : s_cy[m];
  b1.x = hi ? s_cz[m + 16]       : s_cx[m + 16];
  b1.y = hi ? 0.0f               : s_cy[m + 16];

  v8f zc = {};
  v8f d00 = __builtin_amdgcn_wmma_f32_16x16x4_f32(false, a0, false, b0, (short)0, zc, false, false);
  v8f d01 = __builtin_amdgcn_wmma_f32_16x16x4_f32(false, a0, false, b1, (short)0, zc, false, false);
  v8f d10 = __builtin_amdgcn_wmma_f32_16x16x4_f32(false, a1, false, b0, (short)0, zc, false, false);
  v8f d11 = __builtin_amdgcn_wmma_f32_16x16x4_f32(false, a1, false, b1, (short)0, zc, false, false);

  // D layout: VGPR r, lanes 0-15: M=r; lanes 16-31: M=8+r; N = lane&15.
  const int pm = (hi ? 8 : 0);
#pragma unroll
  for (int r = 0; r < 8; ++r) {
    s_dot[w][pm + r][m]           = d00[r];
    s_dot[w][pm + r][m + 16]      = d01[r];
    s_dot[w][pm + r + 16][m]      = d10[r];
    s_dot[w][pm + r + 16][m + 16] = d11[r];
  }
  __syncthreads();

  // Epilogue: lane l owns point (n = block*256 + w*32 + l) -> row l of its wave.
  const float cn = cex * cex + cey * cey + cez * cez;
  float masks = 0.0f, smoothc = 0.0f, seedc = 0.0f;
  float best = 3.4e38f;
  int bestk = 0;
#pragma unroll 4
  for (int k = 0; k < KK; ++k) {
    const float dot = s_dot[w][l][k];
    const float d2  = cn - 2.0f * dot + s_n2[k];
    const float arg = fminf(fmaxf(d2 * s_i2[k], 1e-7f), 100.0f);
    const float p   = expf(-arg);
    const bool  is  = (k == inst);
    masks += is ? arg : (-log1pf(-p));
    if (is) {
      smoothc = fabsf(sig - s_sg[k]) * s_ic[k];
      const float e = sd - p;
      seedc = e * e * s_ic[k];
    }
    if (d2 < best) { best = d2; bestk = k; }
  }
  const float acchit = (bestk == inst) ? 1.0f : 0.0f;

  atomicAdd(&s_acc[0], masks);
  atomicAdd(&s_acc[1], smoothc);
  atomicAdd(&s_acc[2], seedc);
  atomicAdd(&s_acc[3], acchit);
  __syncthreads();
  if (tid < 4) atomicAdd(&ws[WS_ACC + b * 4 + tid], s_acc[tid]);
}

// Final combine: K x K inter-center hinge + scalar loss/acc.
__global__ __launch_bounds__(256) void cl_final(const float* __restrict__ ws,
                                                float* __restrict__ out) {
  __shared__ float inter[BB];
  const int t = threadIdx.x;
  if (t < BB) inter[t] = 0.0f;
  __syncthreads();
  for (int i = t; i < BB * KK * KK; i += 256) {
    const int b  = i >> 10;
    const int r  = i & 1023;
    const int ki = r >> 5;
    const int kj = r & 31;
    if (ki != kj) {
      const float* ci = ws + WS_CEN + (b * KK + ki) * 3;
      const float* cj = ws + WS_CEN + (b * KK + kj) * 3;
      const float dx = ci[0] - cj[0], dy = ci[1] - cj[1], dz = ci[2] - cj[2];
      const float d  = sqrtf(dx * dx + dy * dy + dz * dz);
      const float h  = fmaxf(0.0f, 1.0f - d);
      atomicAdd(&inter[b], h * h);
    }
  }
  __syncthreads();
  if (t == 0) {
    float loss = 0.0f, acc = 0.0f;
    for (int b = 0; b < BB; ++b) {
      const float* a = ws + WS_ACC + b * 4;
      loss += a[0] / ((float)NBP * KK)
            + a[1] / (float)KK
            + a[2] / (float)KK
            + inter[b] / (2.0f * KK * (KK - 1));
      acc += a[3];
    }
    out[0] = loss / (float)BB;
    out[1] = acc / (float)NTOT;
  }
}

extern "C" void kernel_launch(void* const* d_in, const int* in_sizes, int n_in,
                              void* d_out, int out_size, void* d_ws, size_t ws_size,
                              hipStream_t stream) {
  const float* le  = (const float*)d_in[0];  // (N,4)
  const float* se  = (const float*)d_in[1];  // (N,1)
  const float* lab = (const float*)d_in[2];  // (N,6)
  float* ws  = (float*)d_ws;
  float* out = (float*)d_out;
  (void)in_sizes; (void)n_in; (void)out_size; (void)ws_size;

  const int nblocks = NTOT / 256;  // 2048
  cl_zero<<<(WS_TOTAL + 255) / 256, 256, 0, stream>>>(ws);
  cl_pass1<<<nblocks, 256, 0, stream>>>(le, lab, ws);
  cl_finalize<<<1, 128, 0, stream>>>(ws);
  cl_pass2<<<nblocks, 256, 0, stream>>>(le, se, lab, ws);
  cl_final<<<1, 256, 0, stream>>>(ws, out);
}